// GPR_46651934769531
// MI455X (gfx1250) — compile-verified
//
#include <hip/hip_runtime.h>
#include <math.h>

// ---------------------------------------------------------------------------
// GFM / GMCCE mask on MI455X (gfx1250, wave32).
//   K1: fused KNN: double-buffered async-to-LDS candidate staging (ASYNCcnt)
//       + V_WMMA_F32_16X16X4_F32 Gram tiles + register top-10
//   K2: 45-pair triangle loss, fully unrolled, two small sorts
//   K3: per-batch min reduce
//   K4: sigmoid threshold -> {0,1}
// ---------------------------------------------------------------------------

typedef float v2f __attribute__((ext_vector_type(2)));
typedef float v8f __attribute__((ext_vector_type(8)));

#define KNN    10
#define NPAIR  45
#define GEPS   1e-6f
#define SHARPNESS 20.0f

// combinations(range(10), 2), reference COMB order
constexpr int CA[NPAIR] = {0,0,0,0,0,0,0,0,0, 1,1,1,1,1,1,1,1, 2,2,2,2,2,2,2,
                           3,3,3,3,3,3, 4,4,4,4,4, 5,5,5,5, 6,6,6, 7,7, 8};
constexpr int CB[NPAIR] = {1,2,3,4,5,6,7,8,9, 2,3,4,5,6,7,8,9, 3,4,5,6,7,8,9,
                           4,5,6,7,8,9, 5,6,7,8,9, 6,7,8,9, 7,8,9, 8,9, 9};

// async copy: one f32 element per lane, global -> LDS (ASYNCcnt-tracked)
__device__ __forceinline__ void async_copy_f32(unsigned lds_off,
                                               const float* gaddr) {
  asm volatile("global_load_async_to_lds_b32 %0, %1, off"
               :: "v"(lds_off), "v"(gaddr)
               : "memory");
}

// ---------------------------------------------------------------------------
// Kernel 1: KNN. One block = 128 threads = 4 waves; each wave owns 16 query
// points of one batch. 32 candidates per iteration, coordinates staged into
// LDS by the async pipe one tile ahead (double buffered), dot tiles via two
// f32 WMMAs (16x16x4, xyz padded with 0). Lane L keeps a sorted top-10 for
// row (L%16) over column-half (L/16); halves merged at the end, stable by
// (w desc, idx asc) to match jax.lax.top_k tie-breaking.
// ---------------------------------------------------------------------------
__global__ __launch_bounds__(128)
void knn_kernel(const float* __restrict__ src, int* __restrict__ knn_out, int N) {
  __shared__ float cbuf[4][2][3][32];  // [wave][parity][coord][cand]
  __shared__ float ldsD[4][16][33];    // padded: conflict-free rows
  __shared__ float ldsSq[4][32];
  __shared__ float mw[4][2][16][KNN];
  __shared__ int   mi[4][2][16][KNN];

  const int wave = threadIdx.x >> 5;
  const int lane = threadIdx.x & 31;
  const int half = lane >> 4;     // column half this lane scans
  const int l16  = lane & 15;

  const int tilesPerBatch = N / 64;
  const int b  = blockIdx.x / tilesPerBatch;
  const int q0 = (blockIdx.x % tilesPerBatch) * 64 + wave * 16;

  const float* __restrict__ X = src + (size_t)b * 3 * N;
  const float* __restrict__ Y = X + N;
  const float* __restrict__ Z = X + 2 * N;

  // A fragment: 16x4 f32, K = (x,y,z,0).
  // lanes 0-15: VGPR0=K0(x), VGPR1=K1(y); lanes 16-31: VGPR0=K2(z), VGPR1=K3(0)
  const int qr = q0 + l16;
  const float qx = X[qr], qy = Y[qr], qz = Z[qr];
  v2f A;
  A.x = half ? qz : qx;
  A.y = half ? 0.0f : qy;
  const float sqQ = qx * qx + qy * qy + qz * qz;   // |p_row|^2 for row l16

  // generic->LDS offset: low 32 bits of the flat address (LDS aperture rule)
  const unsigned cbase0 = (unsigned)(size_t)&cbuf[wave][0][0][lane];
  const unsigned cbase1 = (unsigned)(size_t)&cbuf[wave][1][0][lane];

  float wb[KNN];
  int   ib[KNN];
#pragma unroll
  for (int s = 0; s < KNN; ++s) { wb[s] = -1.0f; ib[s] = 0x7fffffff; }

  // prologue: stage tile 0 into parity buffer 0 (3 async instructions)
  async_copy_f32(cbase0,       X + lane);
  async_copy_f32(cbase0 + 128, Y + lane);
  async_copy_f32(cbase0 + 256, Z + lane);

  for (int c0 = 0; c0 < N; c0 += 32) {
    const int par = (c0 >> 5) & 1;

    if (c0 + 32 < N) {
      // stage next tile into the other buffer while we consume this one
      const unsigned nb = par ? cbase0 : cbase1;
      const int cn = c0 + 32 + lane;
      async_copy_f32(nb,       X + cn);
      async_copy_f32(nb + 128, Y + cn);
      async_copy_f32(nb + 256, Z + cn);
      asm volatile("s_wait_asynccnt 0x3" ::: "memory");  // 3 oldest (= this tile) done
    } else {
      asm volatile("s_wait_asynccnt 0x0" ::: "memory");
    }

    const float* cb = &cbuf[wave][par][0][0];

    // |p_cand|^2 for the 32 staged candidates
    {
      const float cx = cb[lane], cy = cb[32 + lane], cz = cb[64 + lane];
      ldsSq[wave][lane] = cx * cx + cy * cy + cz * cz;
    }

    // B fragments (4x16) from staged LDS:
    // lanes 0-15: rows K0(x),K1(y); lanes 16-31: K2(z),K3(0)
    const float b0x = cb[l16],      b0y = cb[32 + l16],      b0z = cb[64 + l16];
    const float b1x = cb[16 + l16], b1y = cb[48 + l16],      b1z = cb[80 + l16];
    v2f B0, B1;
    B0.x = half ? b0z : b0x;  B0.y = half ? 0.0f : b0y;
    B1.x = half ? b1z : b1x;  B1.y = half ? 0.0f : b1y;

    v8f Cz = {0.f, 0.f, 0.f, 0.f, 0.f, 0.f, 0.f, 0.f};
    // D[M][Ncol] = dot(query q0+M, candidate)
    v8f D0 = __builtin_amdgcn_wmma_f32_16x16x4_f32(
        false, A, false, B0, (short)0, Cz, false, false);
    v8f D1 = __builtin_amdgcn_wmma_f32_16x16x4_f32(
        false, A, false, B1, (short)0, Cz, false, false);

    // C/D layout: VGPR r holds M = r + 8*(lane>=16), col = lane%16
#pragma unroll
    for (int r = 0; r < 8; ++r) {
      ldsD[wave][r + 8 * half][l16]      = D0[r];
      ldsD[wave][r + 8 * half][16 + l16] = D1[r];
    }
    asm volatile("s_wait_dscnt 0" ::: "memory");   // same-wave LDS handoff

    // scan: lane owns row l16, columns [16*half, 16*half+16)
#pragma unroll 4
    for (int j = 0; j < 16; ++j) {
      const int col = half * 16 + j;
      const float dot = ldsD[wave][l16][col];
      float d2 = sqQ + ldsSq[wave][col] - 2.0f * dot;
      d2 = fmaxf(d2, 0.0f);
      const float w = __expf(-0.5f * d2);
      const int ci = c0 + col;
      if (w > wb[KNN - 1] || (w == wb[KNN - 1] && ci < ib[KNN - 1])) {
        float cw = w; int cidx = ci;
#pragma unroll
        for (int s = 0; s < KNN; ++s) {
          const bool better = (cw > wb[s]) || (cw == wb[s] && cidx < ib[s]);
          const float tw = wb[s]; const int ti = ib[s];
          wb[s] = better ? cw : tw;  ib[s] = better ? cidx : ti;
          cw   = better ? tw : cw;   cidx = better ? ti : cidx;
        }
      }
    }
  }

  // publish both halves' lists, then lanes 0-15 merge (stable order)
#pragma unroll
  for (int s = 0; s < KNN; ++s) {
    mw[wave][half][l16][s] = wb[s];
    mi[wave][half][l16][s] = ib[s];
  }
  asm volatile("s_wait_dscnt 0" ::: "memory");

  if (half == 0) {
    int* outp = knn_out + ((size_t)(b * N + q0 + l16)) * KNN;
    int pa = 0, pb = 0;
#pragma unroll
    for (int s = 0; s < KNN; ++s) {
      const float wa = mw[wave][0][l16][pa]; const int ia = mi[wave][0][l16][pa];
      const float wc = mw[wave][1][l16][pb]; const int ic = mi[wave][1][l16][pb];
      const bool takeA = (wa > wc) || (wa == wc && ia < ic);
      outp[s] = takeA ? ia : ic;
      if (takeA) ++pa; else ++pb;
    }
  }
}

// ---------------------------------------------------------------------------
// Kernel 2: per-point GMCCE loss. One thread per (b,n).
// ---------------------------------------------------------------------------
__device__ inline float tri_area(float ax, float ay, float az,
                                 float bx, float by, float bz) {
  const float cx = ay * bz - az * by;
  const float cy = az * bx - ax * bz;
  const float cz = ax * by - ay * bx;
  const float s  = cx * cx + cy * cy + cz * cz;
  return (s > 0.0f) ? 0.5f * sqrtf(s) : 0.0f;
}

__device__ inline void isort45(float* a) {
  for (int i = 1; i < NPAIR; ++i) {
    const float v = a[i];
    int j = i - 1;
    while (j >= 0 && a[j] > v) { a[j + 1] = a[j]; --j; }
    a[j + 1] = v;
  }
}

__global__ __launch_bounds__(128)
void loss_kernel(const float* __restrict__ src, const float* __restrict__ tgt,
                 const int* __restrict__ knn_in, float* __restrict__ mean_loss,
                 int N) {
  const int gid = blockIdx.x * blockDim.x + threadIdx.x;
  const int b = gid / N;
  const int n = gid % N;

  const float* SX = src + (size_t)b * 3 * N;
  const float* SY = SX + N;
  const float* SZ = SX + 2 * N;
  const float* TX = tgt + (size_t)b * 3 * N;
  const float* TY = TX + N;
  const float* TZ = TX + 2 * N;

  const float p0sx = SX[n], p0sy = SY[n], p0sz = SZ[n];
  const float p0tx = TX[n], p0ty = TY[n], p0tz = TZ[n];

  // neighbor edge vectors (relative to self) kept in registers
  float esx[KNN], esy[KNN], esz[KNN], etx[KNN], ety[KNN], etz[KNN];
#pragma unroll
  for (int k = 0; k < KNN; ++k) {
    const int j = knn_in[(size_t)gid * KNN + k];
    esx[k] = SX[j] - p0sx;  esy[k] = SY[j] - p0sy;  esz[k] = SZ[j] - p0sz;
    etx[k] = TX[j] - p0tx;  ety[k] = TY[j] - p0ty;  etz[k] = TZ[j] - p0tz;
  }

  float loss[NPAIR], reg[NPAIR];
#pragma unroll
  for (int m = 0; m < NPAIR; ++m) {
    const int a = CA[m], c = CB[m];
    const float areaS = tri_area(esx[a], esy[a], esz[a], esx[c], esy[c], esz[c]);
    const float areaT = tri_area(etx[a], ety[a], etz[a], etx[c], ety[c], etz[c]);
    // src_l = [0,0,areaS]; tgt_l = [eps,eps,areaT+eps] (sorted then +eps)
    const float tA   = areaT + GEPS;
    const float dv   = areaS - tA;
    const float ssq  = GEPS * GEPS + GEPS * GEPS + dv * dv;
    const float den  = GEPS + GEPS + (areaS + tA);
    loss[m] = ssq / den;
    reg[m]  = sqrtf(ssq);
  }

  float sl[NPAIR];
  for (int m = 0; m < NPAIR; ++m) sl[m] = loss[m];
  isort45(sl);                                   // sorted_loss

  float total[NPAIR], tt[NPAIR];
  for (int m = 0; m < NPAIR; ++m) {
    total[m] = sl[m] + 0.1f * reg[m];            // NOTE: reg is NOT sorted (matches ref)
    tt[m] = total[m];
  }
  isort45(tt);
  const float thr = 3.0f * tt[(NPAIR - 1) / 2];  // lower median

  float acc = 0.0f;
#pragma unroll
  for (int m = 0; m < KNN; ++m) {                // limited = total[:K]
    float t = total[m];
    t = (t > thr) ? 0.0f : t;
    acc += sqrtf(t + GEPS);
  }
  mean_loss[gid] = acc / 10.0f;
}

// ---------------------------------------------------------------------------
// Kernel 3: per-batch min of mean_loss
// ---------------------------------------------------------------------------
__global__ __launch_bounds__(256)
void min_kernel(const float* __restrict__ mean_loss, float* __restrict__ mn, int N) {
  __shared__ float red[256];
  const int b = blockIdx.x;
  float m = 3.402823466e+38f;
  for (int i = threadIdx.x; i < N; i += 256)
    m = fminf(m, mean_loss[(size_t)b * N + i]);
  red[threadIdx.x] = m;
  __syncthreads();
  for (int s = 128; s > 0; s >>= 1) {
    if (threadIdx.x < s) red[threadIdx.x] = fminf(red[threadIdx.x], red[threadIdx.x + s]);
    __syncthreads();
  }
  if (threadIdx.x == 0) mn[b] = red[0];
}

// ---------------------------------------------------------------------------
// Kernel 4: w = 2*sigmoid(-20*(m - mn)); out = (w > 0.5) ? 1 : 0
// ---------------------------------------------------------------------------
__global__ __launch_bounds__(128)
void final_kernel(const float* __restrict__ mean_loss, const float* __restrict__ mn,
                  float* __restrict__ out, int N) {
  const int gid = blockIdx.x * blockDim.x + threadIdx.x;
  const int b = gid / N;
  const float d = mean_loss[gid] - mn[b];
  const float w = 2.0f / (1.0f + __expf(SHARPNESS * d));
  out[gid] = (w > 0.5f) ? 1.0f : 0.0f;
}

// ---------------------------------------------------------------------------
extern "C" void kernel_launch(void* const* d_in, const int* in_sizes, int n_in,
                              void* d_out, int out_size, void* d_ws, size_t ws_size,
                              hipStream_t stream) {
  const float* src = (const float*)d_in[0];
  const float* tgt = (const float*)d_in[1];
  float* out = (float*)d_out;

  const int B = 4;
  const int N = 8192;   // reference: src/tgt are (4,3,8192) f32

  char* ws = (char*)d_ws;
  int*   knn_buf   = (int*)ws;                                        // B*N*KNN ints
  size_t off       = (size_t)B * N * KNN * sizeof(int);
  float* mean_loss = (float*)(ws + off);                              // B*N floats
  off             += (size_t)B * N * sizeof(float);
  float* mn        = (float*)(ws + off);                              // B floats

  knn_kernel  <<<dim3(B * (N / 64)),   dim3(128), 0, stream>>>(src, knn_buf, N);
  loss_kernel <<<dim3((B * N) / 128),  dim3(128), 0, stream>>>(src, tgt, knn_buf, mean_loss, N);
  min_kernel  <<<dim3(B),              dim3(256), 0, stream>>>(mean_loss, mn, N);
  final_kernel<<<dim3((B * N) / 128),  dim3(128), 0, stream>>>(mean_loss, mn, out, N);
}